// AttentionDigitCaps_71932112273748
// MI455X (gfx1250) — compile-verified
//
#include <hip/hip_runtime.h>

typedef __attribute__((ext_vector_type(2))) float v2f;
typedef __attribute__((ext_vector_type(8))) float v8f;

#define B_SZ   32
#define N_CAPS 2048
#define IL     32
#define C_CH   32
#define L_DIM  32
#define GN     1024   // C*L

// ---------------------------------------------------------------------------
// Stage 1: u[b,j,n] = sum_i x[b,j,i] * W[j,i,n]
// One wave handles (j, group of 4 n-tiles) for ALL 32 batch rows, so each
// W[j] element is loaded exactly once (halves W traffic vs. per-m-tile waves).
// fp32 WMMA 16x16x4, chained 8x across K=32 for exact fp32 numerics.
// Per n-tile only 2 accumulators (16 VGPRs) are live -> no spill pressure;
// the two independent WMMA chains (mt0/mt1) share each B-operand load.
// ---------------------------------------------------------------------------
__global__ __launch_bounds__(256)
void caps_gemm_u(const float* __restrict__ x, const float* __restrict__ W,
                 float* __restrict__ u)
{
  const int lane = threadIdx.x & 31;
  const int wid  = blockIdx.x * 8 + (threadIdx.x >> 5);
  const int ng   = wid & 15;         // 16 groups of 4 n-tiles (64 cols each)
  const int j    = wid >> 4;         // capsule index 0..2047
  const int half = lane >> 4;        // ISA: lanes 16-31 carry K+2 / rows M+8
  const int l16  = lane & 15;

  // A operands (16x4 fp32 per chunk): row = batch, per-lane K = kc*4+2*half+{0,1}
  const float* xr0 = x + ((size_t)l16 * N_CAPS + j) * IL;         // rows 0..15
  const float* xr1 = x + ((size_t)(16 + l16) * N_CAPS + j) * IL;  // rows 16..31
  v2f a0[8], a1[8];
#pragma unroll
  for (int kc = 0; kc < 8; ++kc) {
    const int k = kc * 4 + 2 * half;
    a0[kc].x = xr0[k];  a0[kc].y = xr0[k + 1];
    a1[kc].x = xr1[k];  a1[kc].y = xr1[k + 1];
  }

#pragma unroll
  for (int t = 0; t < 4; ++t) {
    const int col = ng * 64 + t * 16 + l16;
    const float* wc = W + (size_t)j * IL * GN + col;
    v8f acc0 = {};
    v8f acc1 = {};
#pragma unroll
    for (int kc = 0; kc < 8; ++kc) {
      const int k = kc * 4 + 2 * half;
      v2f bm;
      bm.x = wc[(size_t)k * GN];        // B operand 4x16: lane = column
      bm.y = wc[(size_t)(k + 1) * GN];
      acc0 = __builtin_amdgcn_wmma_f32_16x16x4_f32(false, a0[kc], false, bm,
                                                   (short)0, acc0, false, false);
      acc1 = __builtin_amdgcn_wmma_f32_16x16x4_f32(false, a1[kc], false, bm,
                                                   (short)0, acc1, false, false);
    }
    // D layout: lanes 0-15 rows 0..7, lanes 16-31 rows 8..15, N = lane&15
#pragma unroll
    for (int r = 0; r < 8; ++r) {
      const int brow = half * 8 + r;
      u[((size_t)brow * N_CAPS + j) * GN + col]        = acc0[r];
      u[((size_t)(16 + brow) * N_CAPS + j) * GN + col] = acc1[r];
    }
  }
}

// ---------------------------------------------------------------------------
// Stage 2 (x3): fused logit-update + softmax + weighted partial reduction.
// Block = (b, chunk of 64 capsules), 8 waves x 8 capsules. lane = channel c.
// iter==0: c = 1/32 (softmax of zeros) — no logits touched.
// iter==1: b = u.v0, store logits, softmax.
// iter==2: b = blog + u.v1, softmax (no store needed).
// Deterministic: per-chunk partials, no float atomics.
// ---------------------------------------------------------------------------
__global__ __launch_bounds__(256)
void caps_route_partial(const float* __restrict__ u, float* __restrict__ blog,
                        const float* __restrict__ vprev,
                        float* __restrict__ spart, int iter)
{
  __shared__ float vsh[GN];
  __shared__ float sacc[8 * GN];
  const int tid   = threadIdx.x;
  const int lane  = tid & 31;   // channel c
  const int w     = tid >> 5;
  const int chunk = blockIdx.x; // 0..31 (64 capsules each)
  const int b     = blockIdx.y; // 0..31

  if (iter != 0) {
    for (int i = tid; i < GN; i += 256) vsh[i] = vprev[(size_t)b * GN + i];
  }
  for (int i = tid; i < 8 * GN; i += 256) sacc[i] = 0.0f;
  __syncthreads();

  for (int jj = 0; jj < 8; ++jj) {
    const int j = chunk * 64 + w * 8 + jj;
    const float* up = u + (((size_t)b * N_CAPS + j) * C_CH + lane) * L_DIM;
    float ur[32];
    const float4* up4 = reinterpret_cast<const float4*>(up);
#pragma unroll
    for (int q = 0; q < 8; ++q) {
      float4 t4 = up4[q];
      ur[4 * q + 0] = t4.x; ur[4 * q + 1] = t4.y;
      ur[4 * q + 2] = t4.z; ur[4 * q + 3] = t4.w;
    }

    float cch;
    if (iter == 0) {
      cch = 1.0f / 32.0f;
    } else {
      float d = 0.0f;
#pragma unroll
      for (int l = 0; l < 32; ++l) d = fmaf(ur[l], vsh[lane * L_DIM + l], d);
      float logit = d;
      const size_t bi = ((size_t)b * N_CAPS + j) * C_CH + lane;
      if (iter >= 2) logit += blog[bi];
      if (iter == 1) blog[bi] = logit;
      float m = logit;
#pragma unroll
      for (int off = 16; off >= 1; off >>= 1)
        m = fmaxf(m, __shfl_xor(m, off, 32));
      const float e = __expf(logit - m);
      float sum = e;
#pragma unroll
      for (int off = 16; off >= 1; off >>= 1) sum += __shfl_xor(sum, off, 32);
      cch = e / sum;
    }

    float* sl = sacc + w * GN + lane * L_DIM;
#pragma unroll
    for (int l = 0; l < 32; ++l) {
      const int l2 = (l + lane) & 31;   // bank-conflict-free rotation
      sl[l2] += cch * ur[l2];
    }
  }
  __syncthreads();

  for (int i = tid; i < GN; i += 256) {
    float s = 0.0f;
#pragma unroll
    for (int ww = 0; ww < 8; ++ww) s += sacc[ww * GN + i];
    spart[((size_t)b * 32 + chunk) * GN + i] = s;
  }
}

// ---------------------------------------------------------------------------
// Stage 3 (x3): reduce chunk partials + bias, squash -> v
// block = (32 lanes = L dim, 8 channel groups); one block per batch b.
// ---------------------------------------------------------------------------
__global__ __launch_bounds__(256)
void caps_finalize(const float* __restrict__ spart,
                   const float* __restrict__ biases, float* __restrict__ vout)
{
  const int b  = blockIdx.x;
  const int lx = threadIdx.x;   // l
  const int ty = threadIdx.y;   // channel subgroup
  for (int ch = ty; ch < C_CH; ch += 8) {
    const int n = ch * L_DIM + lx;
    float s = 0.0f;
#pragma unroll
    for (int ck = 0; ck < 32; ++ck) s += spart[((size_t)b * 32 + ck) * GN + n];
    s += biases[n];
    float n2 = s * s;
#pragma unroll
    for (int off = 16; off >= 1; off >>= 1) n2 += __shfl_xor(n2, off, 32);
    const float nn = sqrtf(n2);
    const float f  = n2 / (1.0f + n2) / (nn + 1e-7f);
    vout[(size_t)b * GN + n] = s * f;
  }
}

extern "C" void kernel_launch(void* const* d_in, const int* in_sizes, int n_in,
                              void* d_out, int out_size, void* d_ws,
                              size_t ws_size, hipStream_t stream)
{
  (void)in_sizes; (void)n_in; (void)out_size; (void)ws_size;
  const float* x  = (const float*)d_in[0];
  const float* W  = (const float*)d_in[1];
  const float* bi = (const float*)d_in[2];

  float* u     = (float*)d_ws;                          // 64Mi floats (256 MB)
  float* blog  = u + (size_t)B_SZ * N_CAPS * GN;        // 2Mi floats (8 MB)
  float* spart = blog + (size_t)B_SZ * N_CAPS * C_CH;   // 1Mi floats (4 MB)
  float* vws   = spart + (size_t)B_SZ * 32 * GN;        // 32Ki floats (128 KB)

  // 2048 j * 16 n-groups = 32768 waves / 8 waves per block
  caps_gemm_u<<<4096, 256, 0, stream>>>(x, W, u);
  for (int iter = 0; iter < 3; ++iter) {
    caps_route_partial<<<dim3(32, 32), 256, 0, stream>>>(u, blog, vws, spart,
                                                         iter);
    float* vo = (iter == 2) ? (float*)d_out : vws;
    caps_finalize<<<dim3(32), dim3(32, 8), 0, stream>>>(spart, bi, vo);
  }
}